// CompGCNLayer_11905649344577
// MI455X (gfx1250) — compile-verified
//
#include <hip/hip_runtime.h>
#include <hip/hip_bf16.h>

typedef __attribute__((ext_vector_type(2))) float v2f;
typedef __attribute__((ext_vector_type(8))) float v8f;

#define NN     50000      // nodes
#define NPAD   50048      // padded to multiple of 128 (WMMA strips)
#define DD     128        // feature dim == output dim
#define EDIR   800000     // edges per direction
#define ETOT   1600000    // total edges
#define NRELS  474        // 2*237 relation embeddings (loop rel separate)
#define BN_EPS_F 1e-5f

// ---------------------------------------------------------------- degrees
__global__ void __launch_bounds__(256)
degree_kernel(const int* __restrict__ ei, float* __restrict__ deg_in,
              float* __restrict__ deg_rev) {
  const int stride = gridDim.x * blockDim.x;
  for (int e = blockIdx.x * blockDim.x + threadIdx.x; e < ETOT; e += stride) {
    const int dst = ei[e];  // row = edge_index[0][e]
    if (e < EDIR) atomicAdd(&deg_in[dst], 1.0f);
    else          atomicAdd(&deg_rev[dst], 1.0f);
  }
}

// ------------------------------------------------- edge scatter (wave/edge)
__global__ void __launch_bounds__(256)
scatter_kernel(const float* __restrict__ x, const float* __restrict__ rel,
               const int* __restrict__ ei, const int* __restrict__ et,
               float* __restrict__ h_in, float* __restrict__ h_rev) {
  const int lane = threadIdx.x & 31;
  const int wid  = (blockIdx.x * blockDim.x + threadIdx.x) >> 5;
  const int nw   = (gridDim.x * blockDim.x) >> 5;
  for (int e = wid; e < ETOT; e += nw) {
    const int dst = ei[e];          // row
    const int src = ei[ETOT + e];   // col
    const int t   = et[e];
    const float4 xv = ((const float4*)(x + (long)src * DD))[lane];
    const float4 rv = ((const float4*)(rel + (long)t * DD))[lane];
    float* __restrict__ h = (e < EDIR) ? h_in : h_rev;  // uniform per wave
    float* p = h + (long)dst * DD + lane * 4;
    atomicAdd(p + 0, xv.x * rv.x);
    atomicAdd(p + 1, xv.y * rv.y);
    atomicAdd(p + 2, xv.z * rv.z);
    atomicAdd(p + 3, xv.w * rv.w);
  }
}

// ------------------------------------- weight repack: pair-interleave K rows
// Wp[(k>>1)*256 + n*2 + (k&1)] = W[k*128 + n]  -> B fragments become b64 loads
__global__ void __launch_bounds__(256)
repack_kernel(const float* __restrict__ W_in, const float* __restrict__ W_out,
              const float* __restrict__ W_loop, const float* __restrict__ W_rel,
              float* __restrict__ Wp) {
  const int i = blockIdx.x * blockDim.x + threadIdx.x;
  if (i < 4 * DD * DD) {
    const int w   = i >> 14;        // which weight
    const int idx = i & (DD * DD - 1);
    const int k   = idx >> 7;
    const int n   = idx & (DD - 1);
    const float* src = (w == 0) ? W_in : (w == 1) ? W_out
                     : (w == 2) ? W_loop : W_rel;
    Wp[w * DD * DD + (k >> 1) * 256 + n * 2 + (k & 1)] = src[idx];
  }
}

// --------------------------------------- one K=128 WMMA pass for a 16x128 strip
// A-frag (16x4 f32): lanes 0-15 hold M=lane,K={k,k+1}; lanes 16-31 hold K={k+2,k+3}
// B-frag (4x16 f32): same K split across half-waves; packed layout -> b64 loads.
__device__ __forceinline__ void
kpass(v8f acc[8], const float* __restrict__ Arow, float rowscale,
      const float* __restrict__ colscale, const float* __restrict__ Wp, int lane) {
  const int l15 = lane & 15;
  const int hh  = lane >> 4;
#pragma unroll 4
  for (int k = 0; k < DD; k += 4) {
    const int kk = k + hh * 2;           // even -> 8B aligned
    v2f a = *(const v2f*)(Arow + kk);
    if (colscale) {
      const v2f c = *(const v2f*)(colscale + kk);
      a.x *= c.x; a.y *= c.y;
    }
    a.x *= rowscale; a.y *= rowscale;
    const float* wb = Wp + (kk >> 1) * 256;
#pragma unroll
    for (int t = 0; t < 8; ++t) {
      const v2f b = *(const v2f*)(wb + (t * 16 + l15) * 2);
      acc[t] = __builtin_amdgcn_wmma_f32_16x16x4_f32(
          false, a, false, b, (short)0, acc[t], false, false);
    }
  }
}

// -------------------------- main fused GEMM: 3 passes + bias + BN stat atomics
__global__ void __launch_bounds__(256)
main_gemm_kernel(const float* __restrict__ x, const float* __restrict__ loop_rel,
                 const float* __restrict__ Wp,  // packed: [W_in][W_out][W_loop][W_rel]
                 const float* __restrict__ bias,
                 const float* __restrict__ h_in, const float* __restrict__ h_rev,
                 const float* __restrict__ deg_in, const float* __restrict__ deg_rev,
                 float* __restrict__ out, float* __restrict__ colsum,
                 float* __restrict__ colsumsq) {
  const int lane  = threadIdx.x & 31;
  const int wave  = threadIdx.x >> 5;
  const int strip = blockIdx.x * 8 + wave;
  const int row0  = strip * 16;
  if (row0 >= NPAD) return;
  const int l15 = lane & 15;
  const int hh  = lane >> 4;
  const int m   = row0 + l15;  // A-row owned by this lane (h/deg buffers padded)

  const v8f zero = {0.f, 0.f, 0.f, 0.f, 0.f, 0.f, 0.f, 0.f};
  v8f acc[8];
#pragma unroll
  for (int t = 0; t < 8; ++t) acc[t] = zero;

  // pass 1: (h_in * inv_deg_in) @ W_in
  float dg  = deg_in[m];
  float inv = dg > 0.0f ? 1.0f / dg : 0.0f;
  kpass(acc, h_in + (long)m * DD, inv, nullptr, Wp, lane);
  // pass 2: (h_rev * inv_deg_rev) @ W_out
  dg  = deg_rev[m];
  inv = dg > 0.0f ? 1.0f / dg : 0.0f;
  kpass(acc, h_rev + (long)m * DD, inv, nullptr, Wp + DD * DD, lane);
  // pass 3: (x ⊙ loop_rel) @ W_loop  (x not padded -> clamp row, mask scale)
  const int   mc   = m < NN ? m : NN - 1;
  const float mask = m < NN ? 1.0f : 0.0f;
  kpass(acc, x + (long)mc * DD, mask, loop_rel, Wp + 2 * DD * DD, lane);

  // epilogue: /3 + bias, store pre-BN, accumulate column sum / sumsq
  const float third = 1.0f / 3.0f;
#pragma unroll
  for (int t = 0; t < 8; ++t) {
    const int n = t * 16 + l15;
    const float bsn = bias[n];
    float s = 0.0f, sq = 0.0f;
#pragma unroll
    for (int r = 0; r < 8; ++r) {
      const int mr = row0 + r + hh * 8;  // C/D layout: VGPR r -> M=r (+8 upper half)
      const float v = acc[t][r] * third + bsn;
      if (mr < NN) {
        out[(long)mr * DD + n] = v;
        s += v; sq += v * v;
      }
    }
    atomicAdd(&colsum[n], s);
    atomicAdd(&colsumsq[n], sq);
  }
}

// ------------------------------------------------ rel_out = (rel @ W_rel)[:-1]
__global__ void __launch_bounds__(256)
rel_gemm_kernel(const float* __restrict__ rel_embed, const float* __restrict__ Wp_rel,
                float* __restrict__ out_rel) {
  const int lane  = threadIdx.x & 31;
  const int wave  = threadIdx.x >> 5;
  const int row0  = (blockIdx.x * 8 + wave) * 16;
  const int l15 = lane & 15;
  const int hh  = lane >> 4;
  const int m   = row0 + l15;

  const v8f zero = {0.f, 0.f, 0.f, 0.f, 0.f, 0.f, 0.f, 0.f};
  v8f acc[8];
#pragma unroll
  for (int t = 0; t < 8; ++t) acc[t] = zero;

  const int   mc   = m < NRELS ? m : NRELS - 1;
  const float mask = m < NRELS ? 1.0f : 0.0f;
  kpass(acc, rel_embed + (long)mc * DD, mask, nullptr, Wp_rel, lane);

#pragma unroll
  for (int t = 0; t < 8; ++t) {
    const int n = t * 16 + l15;
#pragma unroll
    for (int r = 0; r < 8; ++r) {
      const int mr = row0 + r + hh * 8;
      if (mr < NRELS) out_rel[(long)mr * DD + n] = acc[t][r];
    }
  }
}

// ---------------------------------------------------------------- batch norm
__global__ void bn_finalize_kernel(const float* __restrict__ colsum,
                                   const float* __restrict__ colsumsq,
                                   const float* __restrict__ gamma,
                                   const float* __restrict__ beta,
                                   float* __restrict__ scale,
                                   float* __restrict__ shift) {
  const int n = threadIdx.x;
  if (n < DD) {
    const float mean = colsum[n] * (1.0f / (float)NN);
    const float var  = colsumsq[n] * (1.0f / (float)NN) - mean * mean;
    const float s    = rsqrtf(var + BN_EPS_F) * gamma[n];
    scale[n] = s;
    shift[n] = beta[n] - mean * s;
  }
}

__global__ void __launch_bounds__(256)
bn_apply_kernel(float* __restrict__ out, const float* __restrict__ scale,
                const float* __restrict__ shift) {
  float4* o4 = (float4*)out;
  const float4* s4 = (const float4*)scale;
  const float4* b4 = (const float4*)shift;
  const long total = (long)NN * DD / 4;
  const long stride = (long)gridDim.x * blockDim.x;
  for (long i = blockIdx.x * (long)blockDim.x + threadIdx.x; i < total; i += stride) {
    const int c = (int)(i & 31);  // 32 float4 per 128-wide row
    float4 v = o4[i];
    const float4 s = s4[c];
    const float4 b = b4[c];
    v.x = v.x * s.x + b.x;
    v.y = v.y * s.y + b.y;
    v.z = v.z * s.z + b.z;
    v.w = v.w * s.w + b.w;
    o4[i] = v;
  }
}

// ---------------------------------------------------------------- launcher
extern "C" void kernel_launch(void* const* d_in, const int* in_sizes, int n_in,
                              void* d_out, int out_size, void* d_ws, size_t ws_size,
                              hipStream_t stream) {
  const float* x         = (const float*)d_in[0];
  const float* rel_embed = (const float*)d_in[1];
  const int*   edge_idx  = (const int*)d_in[2];
  const int*   edge_type = (const int*)d_in[3];
  const float* w_in      = (const float*)d_in[4];
  const float* w_out     = (const float*)d_in[5];
  const float* w_rel     = (const float*)d_in[6];
  const float* w_loop    = (const float*)d_in[7];
  const float* loop_rel  = (const float*)d_in[8];
  const float* bias      = (const float*)d_in[9];
  const float* gamma     = (const float*)d_in[10];
  const float* beta      = (const float*)d_in[11];

  float* ws       = (float*)d_ws;
  float* h_in     = ws;                               // NPAD*128
  float* h_rev    = h_in  + (size_t)NPAD * DD;        // NPAD*128
  float* deg_in   = h_rev + (size_t)NPAD * DD;        // NPAD
  float* deg_rev  = deg_in + NPAD;                    // NPAD
  float* colsum   = deg_rev + NPAD;                   // 128
  float* colsumsq = colsum + DD;                      // 128
  float* bnscale  = colsumsq + DD;                    // 128
  float* bnshift  = bnscale + DD;                     // 128
  float* Wpack    = bnshift + DD;                     // 4 * 128*128 (packed weights)

  const size_t zbytes =
      ((size_t)2 * NPAD * DD + 2 * NPAD + 2 * DD) * sizeof(float);
  hipMemsetAsync(d_ws, 0, zbytes, stream);

  repack_kernel<<<(4 * DD * DD + 255) / 256, 256, 0, stream>>>(
      w_in, w_out, w_loop, w_rel, Wpack);
  degree_kernel<<<1024, 256, 0, stream>>>(edge_idx, deg_in, deg_rev);
  scatter_kernel<<<4096, 256, 0, stream>>>(x, rel_embed, edge_idx, edge_type,
                                           h_in, h_rev);

  float* out     = (float*)d_out;
  float* out_rel = out + (size_t)NN * DD;

  main_gemm_kernel<<<NPAD / 128, 256, 0, stream>>>(
      x, loop_rel, Wpack, bias, h_in, h_rev, deg_in, deg_rev,
      out, colsum, colsumsq);
  rel_gemm_kernel<<<4, 256, 0, stream>>>(rel_embed, Wpack + 3 * DD * DD, out_rel);
  bn_finalize_kernel<<<1, 128, 0, stream>>>(colsum, colsumsq, gamma, beta,
                                            bnscale, bnshift);
  bn_apply_kernel<<<2048, 256, 0, stream>>>(out, bnscale, bnshift);
}